// OtsuThreshold_21912923144366
// MI455X (gfx1250) — compile-verified
//
#include <hip/hip_runtime.h>

// Otsu threshold, MI455X (gfx1250, wave32).
//   Pass 1: 256-bin count histogram (LDS ds_add atomics, global atomic flush).
//   Pass 2: single wave32 solver. Double cumulative sum over the 256 bins done
//           as two 16x16x16 f32 GEMM chains on the WMMA unit
//           (V_WMMA_F32_16X16X4_F32, 16 instructions total), then per-lane
//           inter-class variance + wave argmax reduction.
//   Pass 3: binarize (b128 loads, image is L2-resident after pass 1).
// HBM traffic ~128-192 MiB -> ~6-8 us at 23.3 TB/s; everything else is noise.

#define HW_N (4096 * 4096)   // 2^24 pixels (exactly representable in f32)
#define N4   (HW_N / 4)

typedef __attribute__((ext_vector_type(2))) float v2f;
typedef __attribute__((ext_vector_type(8))) float v8f;

__global__ __launch_bounds__(256) void otsu_zero(unsigned* __restrict__ gbins) {
    gbins[threadIdx.x] = 0u;
}

__global__ __launch_bounds__(256) void otsu_hist(const int* __restrict__ img,
                                                 unsigned* __restrict__ gbins) {
    __shared__ unsigned bins[256];
    bins[threadIdx.x] = 0u;
    __syncthreads();

    const int stride = gridDim.x * blockDim.x;
    const int4* img4 = (const int4*)img;
    for (int i = blockIdx.x * blockDim.x + threadIdx.x; i < N4; i += stride) {
        int4 p = img4[i];                 // global_load_b128
        atomicAdd(&bins[p.x & 255], 1u);  // ds_add_u32 (no return)
        atomicAdd(&bins[p.y & 255], 1u);
        atomicAdd(&bins[p.z & 255], 1u);
        atomicAdd(&bins[p.w & 255], 1u);
    }
    __syncthreads();
    atomicAdd(&gbins[threadIdx.x], bins[threadIdx.x]);  // global_atomic_add_u32
}

// One wave32. Computes, for every threshold t in [0,254]:
//   num_b(t) = cumsum(cnt)[t], sum_b(t) = cumsum(v*cnt[v])[t]
// via WMMA f32 GEMMs, then the Otsu variance and its argmax.
__global__ __launch_bounds__(32) void otsu_solve(const unsigned* __restrict__ gbins,
                                                 int* __restrict__ out_thresh,
                                                 int* __restrict__ ws_thresh) {
    const int lane = threadIdx.x;      // 0..31
    const int hi   = lane >> 4;        // 0 or 1 (wave half)
    const int m    = lane & 15;        // A-row / B-col index

    // A-layout (16x4 f32, chunk q covers K=4q..4q+3):
    //   vgpr j, lanes 0-15: K=4q+j ; lanes 16-31: K=4q+2+j  -> k = 4q + 2*hi + j
    // B-layout is the transpose-analog: lane's row k = 4q + 2*hi + j, col n = m.
    v2f hA[4], sA[4], uB[4], lA[4];
#pragma unroll
    for (int q = 0; q < 4; ++q) {
#pragma unroll
        for (int j = 0; j < 2; ++j) {
            const int k   = 4 * q + 2 * hi + j;
            const int bin = 16 * m + k;
            const float f = (float)gbins[bin];
            hA[q][j] = f;                          // H[m][k]  (counts)
            sA[q][j] = f * (float)bin;             // H'[m][k] (value-weighted)
            uB[q][j] = (k <= m) ? 1.0f : 0.0f;     // U[k][n], inclusive upper-tri
            lA[q][j] = (k <  m) ? 1.0f : 0.0f;     // Lstrict[m][k]
        }
    }

    // Chain 1: Term1 = H x U  (within-group inclusive prefix sums)
    v8f c  = {0.f, 0.f, 0.f, 0.f, 0.f, 0.f, 0.f, 0.f};
    v8f cs = {0.f, 0.f, 0.f, 0.f, 0.f, 0.f, 0.f, 0.f};
#pragma unroll
    for (int q = 0; q < 4; ++q)
        c = __builtin_amdgcn_wmma_f32_16x16x4_f32(false, hA[q], false, uB[q],
                                                  (short)0, c, false, false);
#pragma unroll
    for (int q = 0; q < 4; ++q)
        cs = __builtin_amdgcn_wmma_f32_16x16x4_f32(false, sA[q], false, uB[q],
                                                   (short)0, cs, false, false);

    // Row sums = column 15 of Term1: lane 15 holds rows 0..7, lane 31 rows 8..15.
    __shared__ float rsC[16], rsS[16];
    if (m == 15) {
#pragma unroll
        for (int v = 0; v < 8; ++v) { rsC[8 * hi + v] = c[v]; rsS[8 * hi + v] = cs[v]; }
    }
    __syncthreads();   // EXEC reconverges to all-1s before the next WMMAs

    // Chain 2: C += Lstrict x Rmat, Rmat[k][n] = rowsum(k) broadcast across n.
    v2f rBc[4], rBs[4];
#pragma unroll
    for (int q = 0; q < 4; ++q) {
#pragma unroll
        for (int j = 0; j < 2; ++j) {
            const int k = 4 * q + 2 * hi + j;
            rBc[q][j] = rsC[k];
            rBs[q][j] = rsS[k];
        }
    }
#pragma unroll
    for (int q = 0; q < 4; ++q)
        c = __builtin_amdgcn_wmma_f32_16x16x4_f32(false, lA[q], false, rBc[q],
                                                  (short)0, c, false, false);
#pragma unroll
    for (int q = 0; q < 4; ++q)
        cs = __builtin_amdgcn_wmma_f32_16x16x4_f32(false, lA[q], false, rBs[q],
                                                   (short)0, cs, false, false);

    // Now element (r, n) of c/cs (r = vgpr + 8*hi, n = m) is the inclusive
    // cumsum at bin t = 16*r + n. Total value-sum = cumsum at bin 255
    // = vgpr 7 of lane 31.
    const float totS = __shfl(cs[7], 31, 32);

    float bestv = -2.0f;
    int   bestt = 0x7fffffff;
#pragma unroll
    for (int v = 0; v < 8; ++v) {
        const int   t  = 16 * (v + 8 * hi) + m;
        const float nb = c[v];
        const float nw = (float)HW_N - nb;
        float var = -1.0f;
        if (t <= 254 && nb > 0.0f && nw > 0.0f) {
            const float mb = cs[v] / nb;
            const float mw = (totS - cs[v]) / nw;
            const float d  = mb - mw;
            var = nb * nw * d * d;
        }
        if (var > bestv || (var == bestv && t < bestt)) { bestv = var; bestt = t; }
    }
#pragma unroll
    for (int off = 16; off > 0; off >>= 1) {
        const float ov = __shfl_xor(bestv, off, 32);
        const int   ot = __shfl_xor(bestt, off, 32);
        if (ov > bestv || (ov == bestv && ot < bestt)) { bestv = ov; bestt = ot; }
    }
    if (lane == 0) {
        out_thresh[0] = bestt;   // tuple element 0 of d_out (int32)
        ws_thresh[0]  = bestt;   // for the binarize pass
    }
}

__global__ __launch_bounds__(256) void otsu_binarize(const int* __restrict__ img,
                                                     const int* __restrict__ ws_thresh,
                                                     int* __restrict__ outimg) {
    const int t = ws_thresh[0];               // uniform -> scalar load
    const int stride = gridDim.x * blockDim.x;
    const int4* img4 = (const int4*)img;
    for (int i = blockIdx.x * blockDim.x + threadIdx.x; i < N4; i += stride) {
        const int4 p = img4[i];               // global_load_b128 (L2-resident)
        const int  j = 4 * i;                 // outimg is d_out+1 -> b32 stores
        outimg[j + 0] = (p.x <= t) ? 0 : 256;
        outimg[j + 1] = (p.y <= t) ? 0 : 256;
        outimg[j + 2] = (p.z <= t) ? 0 : 256;
        outimg[j + 3] = (p.w <= t) ? 0 : 256;
    }
}

extern "C" void kernel_launch(void* const* d_in, const int* in_sizes, int n_in,
                              void* d_out, int out_size, void* d_ws, size_t ws_size,
                              hipStream_t stream) {
    (void)in_sizes; (void)n_in; (void)out_size; (void)ws_size;
    const int* img  = (const int*)d_in[0];
    int*       out  = (int*)d_out;                         // [0]=thresh, [1..]=bin_img
    unsigned*  bins = (unsigned*)d_ws;                     // 256 x u32 counts
    int*       thr  = (int*)((char*)d_ws + 256 * sizeof(unsigned));

    otsu_zero<<<1, 256, 0, stream>>>(bins);                // ws is not re-poisoned: zero each call
    otsu_hist<<<4096, 256, 0, stream>>>(img, bins);
    otsu_solve<<<1, 32, 0, stream>>>(bins, out, thr);
    otsu_binarize<<<4096, 256, 0, stream>>>(img, thr, out + 1);
}